// GNN_36077725287095
// MI455X (gfx1250) — compile-verified
//
#include <hip/hip_runtime.h>
#include <hip/hip_bf16.h>

#define N_NODES 100000
#define N_EDGES 3200000
#define IN_DIM  64
#define HID_DIM 32
#define OUT_DIM 4

typedef __attribute__((ext_vector_type(2))) float v2f;
typedef __attribute__((ext_vector_type(8))) float v8f;

// ---------------- init: deg=1 (self loop), zero agg1 and out ----------------
__global__ void gcn_init(float* __restrict__ deg, float* __restrict__ agg1,
                         float* __restrict__ out) {
  int t = blockIdx.x * blockDim.x + threadIdx.x;
  if (t < N_NODES) deg[t] = 1.0f;                 // self-loop contribution
  if (t < N_NODES * HID_DIM) agg1[t] = 0.0f;
  if (t < N_NODES * OUT_DIM) out[t] = 0.0f;
}

// ---------------- degree accumulation over edges ----------------
__global__ void gcn_degree(const int* __restrict__ dst, float* __restrict__ deg) {
  int e = blockIdx.x * blockDim.x + threadIdx.x;
  if (e < N_EDGES) {
    __builtin_prefetch(dst + e + 4096, 0, 0);
    atomicAdd(&deg[dst[e]], 1.0f);
  }
}

// ---------------- dinv = rsqrt(deg) (deg >= 1 always) ----------------
__global__ void gcn_dinv(const float* __restrict__ deg, float* __restrict__ dinv) {
  int i = blockIdx.x * blockDim.x + threadIdx.x;
  if (i < N_NODES) dinv[i] = rsqrtf(deg[i]);
}

// ---------------- GEMM1: xw[N,32] = x[N,64] @ W1[64,32] via WMMA f32 ----------------
// One wave -> 16 rows x 32 cols (two 16x16 tiles). 8 waves/block -> 128 rows/block.
__global__ __launch_bounds__(256) void gcn_gemm1(const float* __restrict__ x,
                                                 const float* __restrict__ W,
                                                 float* __restrict__ xw) {
  const int lane = threadIdx.x & 31;
  const int wave = threadIdx.x >> 5;
  const int m    = lane & 15;   // row/col within tile
  const int half = lane >> 4;   // K-pair select per ISA f32 A/B layout
  const int row0 = blockIdx.x * 128 + wave * 16;

  int row = row0 + m;
  if (row >= N_NODES) row = N_NODES - 1;          // clamp loads; EXEC stays all-1
  const float* xr = x + (size_t)row * IN_DIM + 2 * half;

  v8f c0 = {};
  v8f c1 = {};
#pragma unroll
  for (int ks = 0; ks < IN_DIM / 4; ++ks) {
    const int k = ks * 4;
    // A fragment 16x4: lanes 0-15 -> K = k,k+1 ; lanes 16-31 -> K = k+2,k+3
    v2f a = *(const v2f*)(xr + k);
    // B fragment 4x16: VGPR0 = K row (k+2*half), VGPR1 = K row (k+2*half+1)
    const float* wr = W + (size_t)(k + 2 * half) * HID_DIM + m;
    v2f b0, b1;
    b0.x = wr[0];        b0.y = wr[HID_DIM];        // cols 0..15
    b1.x = wr[16];       b1.y = wr[HID_DIM + 16];   // cols 16..31
    c0 = __builtin_amdgcn_wmma_f32_16x16x4_f32(false, a, false, b0, (short)0, c0, false, false);
    c1 = __builtin_amdgcn_wmma_f32_16x16x4_f32(false, a, false, b1, (short)0, c1, false, false);
  }

#pragma unroll
  for (int v = 0; v < 8; ++v) {
    const int orow = row0 + v + 8 * half;          // C/D layout: lanes16-31 -> M+8
    if (orow < N_NODES) {
      xw[(size_t)orow * HID_DIM + m]      = c0[v];
      xw[(size_t)orow * HID_DIM + 16 + m] = c1[v];
    }
  }
}

// ---------------- layer-1 aggregation: 8 threads per edge, float4 strips ----------------
__global__ __launch_bounds__(256) void gcn_agg1(const int* __restrict__ src,
                                                const int* __restrict__ dst,
                                                const float* __restrict__ dinv,
                                                const float* __restrict__ xw,
                                                float* __restrict__ agg) {
  int t = blockIdx.x * blockDim.x + threadIdx.x;
  int e = t >> 3;
  if (e >= N_EDGES) return;
  int j0 = (t & 7) * 4;
  __builtin_prefetch(src + e + 2048, 0, 0);
  __builtin_prefetch(dst + e + 2048, 0, 0);
  int s = src[e];
  int d = dst[e];
  float norm = dinv[s] * dinv[d];
  float4 v = *(const float4*)(xw + (size_t)s * HID_DIM + j0);
  float* ap = agg + (size_t)d * HID_DIM + j0;
  atomicAdd(ap + 0, norm * v.x);
  atomicAdd(ap + 1, norm * v.y);
  atomicAdd(ap + 2, norm * v.z);
  atomicAdd(ap + 3, norm * v.w);
}

// ---------------- finalize layer 1: self-loop + bias + ReLU (in place) ----------------
__global__ void gcn_fin1(float* __restrict__ agg, const float* __restrict__ xw,
                         const float* __restrict__ dinv, const float* __restrict__ b1) {
  int t = blockIdx.x * blockDim.x + threadIdx.x;
  if (t >= N_NODES * HID_DIM) return;
  int i = t >> 5;
  int j = t & 31;
  float di = dinv[i];
  float val = agg[t] + di * di * xw[t] + b1[j];
  agg[t] = fmaxf(val, 0.0f);                       // agg becomes h
}

// ---------------- GEMM2: hw[N,4] = h[N,32] @ W2[32,4] via WMMA f32 ----------------
__global__ __launch_bounds__(256) void gcn_gemm2(const float* __restrict__ h,
                                                 const float* __restrict__ W2,
                                                 float* __restrict__ hw) {
  const int lane = threadIdx.x & 31;
  const int wave = threadIdx.x >> 5;
  const int m    = lane & 15;
  const int half = lane >> 4;
  const int row0 = blockIdx.x * 128 + wave * 16;

  int row = row0 + m;
  if (row >= N_NODES) row = N_NODES - 1;
  const float* hr = h + (size_t)row * HID_DIM + 2 * half;

  v8f c = {};
#pragma unroll
  for (int ks = 0; ks < HID_DIM / 4; ++ks) {
    const int k = ks * 4;
    v2f a = *(const v2f*)(hr + k);
    const int kk = k + 2 * half;
    v2f b;
    b.x = (m < OUT_DIM) ? W2[kk * OUT_DIM + m] : 0.0f;
    b.y = (m < OUT_DIM) ? W2[(kk + 1) * OUT_DIM + m] : 0.0f;
    c = __builtin_amdgcn_wmma_f32_16x16x4_f32(false, a, false, b, (short)0, c, false, false);
  }

#pragma unroll
  for (int v = 0; v < 8; ++v) {
    const int orow = row0 + v + 8 * half;
    if (orow < N_NODES && m < OUT_DIM) {
      hw[(size_t)orow * OUT_DIM + m] = c[v];
    }
  }
}

// ---------------- layer-2 aggregation: 1 thread per edge, into d_out ----------------
__global__ __launch_bounds__(256) void gcn_agg2(const int* __restrict__ src,
                                                const int* __restrict__ dst,
                                                const float* __restrict__ dinv,
                                                const float* __restrict__ hw,
                                                float* __restrict__ out) {
  int e = blockIdx.x * blockDim.x + threadIdx.x;
  if (e >= N_EDGES) return;
  __builtin_prefetch(src + e + 2048, 0, 0);
  __builtin_prefetch(dst + e + 2048, 0, 0);
  int s = src[e];
  int d = dst[e];
  float norm = dinv[s] * dinv[d];
  float4 v = *(const float4*)(hw + (size_t)s * OUT_DIM);
  float* op = out + (size_t)d * OUT_DIM;
  atomicAdd(op + 0, norm * v.x);
  atomicAdd(op + 1, norm * v.y);
  atomicAdd(op + 2, norm * v.z);
  atomicAdd(op + 3, norm * v.w);
}

// ---------------- finalize layer 2: self-loop + bias (in place on d_out) ----------------
__global__ void gcn_fin2(float* __restrict__ out, const float* __restrict__ hw,
                         const float* __restrict__ dinv, const float* __restrict__ b2) {
  int t = blockIdx.x * blockDim.x + threadIdx.x;
  if (t >= N_NODES * OUT_DIM) return;
  int i = t >> 2;
  int j = t & 3;
  float di = dinv[i];
  out[t] = out[t] + di * di * hw[t] + b2[j];
}

extern "C" void kernel_launch(void* const* d_in, const int* in_sizes, int n_in,
                              void* d_out, int out_size, void* d_ws, size_t ws_size,
                              hipStream_t stream) {
  (void)in_sizes; (void)n_in; (void)out_size; (void)ws_size;

  const float* x  = (const float*)d_in[0];
  const int*   ei = (const int*)d_in[1];       // [2, E]: row 0 = src, row 1 = dst
  const float* W1 = (const float*)d_in[2];
  const float* b1 = (const float*)d_in[3];
  const float* W2 = (const float*)d_in[4];
  const float* b2 = (const float*)d_in[5];
  float* out = (float*)d_out;

  const int* src = ei;
  const int* dst = ei + N_EDGES;

  // workspace carve-up (floats): deg | dinv | xw | agg1/h | hw
  float* ws   = (float*)d_ws;
  float* deg  = ws;
  float* dinv = deg  + N_NODES;
  float* xw   = dinv + N_NODES;                 // N*32
  float* aggh = xw   + (size_t)N_NODES * HID_DIM; // N*32 (agg1, then h in place)
  float* hw   = aggh + (size_t)N_NODES * HID_DIM; // N*4

  const int T = 256;
  const int initN   = N_NODES * HID_DIM;                 // covers all three init ranges
  const int gemmBlk = (N_NODES + 127) / 128;

  gcn_init  <<<(initN + T - 1) / T, T, 0, stream>>>(deg, aggh, out);
  gcn_degree<<<(N_EDGES + T - 1) / T, T, 0, stream>>>(dst, deg);
  gcn_dinv  <<<(N_NODES + T - 1) / T, T, 0, stream>>>(deg, dinv);

  gcn_gemm1 <<<gemmBlk, T, 0, stream>>>(x, W1, xw);
  gcn_agg1  <<<(N_EDGES * 8 + T - 1) / T, T, 0, stream>>>(src, dst, dinv, xw, aggh);
  gcn_fin1  <<<(N_NODES * HID_DIM + T - 1) / T, T, 0, stream>>>(aggh, xw, dinv, b1);

  gcn_gemm2 <<<gemmBlk, T, 0, stream>>>(aggh, W2, hw);
  gcn_agg2  <<<(N_EDGES + T - 1) / T, T, 0, stream>>>(src, dst, dinv, hw, out);
  gcn_fin2  <<<(N_NODES * OUT_DIM + T - 1) / T, T, 0, stream>>>(out, hw, dinv, b2);
}